// SparseInverseConv3d_14568529068664
// MI455X (gfx1250) — compile-verified
//
#include <hip/hip_runtime.h>

typedef __attribute__((ext_vector_type(2))) float v2f;
typedef __attribute__((ext_vector_type(8))) float v8f;

#define CIN  64
#define COUT 32
#define WAVES 8
#define TILES_PER_WAVE 4
#define PAIRS_PER_BLOCK (WAVES * TILES_PER_WAVE * 16) // 512

// ---------------------------------------------------------------------------
// out[n][c] = bias[c]  (init before scatter-add; rerun every launch)
// ---------------------------------------------------------------------------
__global__ __launch_bounds__(256)
void bias_init_kernel(float* __restrict__ out, const float* __restrict__ bias,
                      int total) {
    int i = blockIdx.x * blockDim.x + threadIdx.x;
    if (i < total) out[i] = bias[i & (COUT - 1)];
}

// ---------------------------------------------------------------------------
// Per kernel-offset k: [P,64] x [64,32] -> scatter-add into out.
// One wave = 16 pairs x 32 Cout, accumulated with V_WMMA_F32_16X16X4_F32.
// ---------------------------------------------------------------------------
__global__ __launch_bounds__(256)
void spconv_wmma_kernel(const float* __restrict__ feats,
                        const float* __restrict__ weight,
                        const int*   __restrict__ in_idx,
                        const int*   __restrict__ out_idx,
                        float*       __restrict__ out,
                        int P) {
    const int k    = blockIdx.y;
    const int lane = threadIdx.x & 31;
    const int wave = threadIdx.x >> 5;
    const int half = lane >> 4;   // 0: K+{0,1} / M 0..7 ; 1: K+{2,3} / M 8..15
    const int l16  = lane & 15;   // A: row M ; B: col N ; D: col N

    // ---- B fragments: weight[k] register-resident (16 K-chunks x 2 N-tiles)
    const float* wk = weight + (size_t)k * (CIN * COUT);
    v2f bfrag[16][2];
#pragma unroll
    for (int kk = 0; kk < 16; ++kk) {
        const int kr = kk * 4 + half * 2;
#pragma unroll
        for (int nt = 0; nt < 2; ++nt) {
            const int col = l16 + nt * 16;
            v2f b;
            b.x = wk[(kr + 0) * COUT + col];
            b.y = wk[(kr + 1) * COUT + col];
            bfrag[kk][nt] = b;
        }
    }

    const int* inik  = in_idx  + (size_t)k * P;
    const int* outik = out_idx + (size_t)k * P;

    for (int t = 0; t < TILES_PER_WAVE; ++t) {
        const int pbase = blockIdx.x * PAIRS_PER_BLOCK
                        + (wave * TILES_PER_WAVE + t) * 16;
        if (pbase >= P) break;  // wave-uniform

        // ---- gather this lane's half-row of feats (clamped: EXEC stays full)
        const int prow  = pbase + l16;
        const int prowc = prow < P ? prow : P - 1;
        const int ridx  = inik[prowc];
        const float* arow = feats + (size_t)ridx * CIN;

        // ---- prefetch next tile's gathered rows (global_prefetch_b8)
        if (t + 1 < TILES_PER_WAVE) {
            const int pn  = pbase + 16 + l16;
            const int pnc = pn < P ? pn : P - 1;
            const int rn  = inik[pnc];
            __builtin_prefetch(feats + (size_t)rn * CIN + half * 32, 0, 0);
        }

        // A fragments: lane m = l16, K = 4*kk + 2*half + {0,1}
        v2f afrag[16];
#pragma unroll
        for (int kk = 0; kk < 16; ++kk)
            afrag[kk] = *(const v2f*)(arow + kk * 4 + half * 2);

        v8f acc0 = {};
        v8f acc1 = {};
#pragma unroll
        for (int kk = 0; kk < 16; ++kk) {
            acc0 = __builtin_amdgcn_wmma_f32_16x16x4_f32(
                false, afrag[kk], false, bfrag[kk][0], (short)0, acc0, false, false);
            acc1 = __builtin_amdgcn_wmma_f32_16x16x4_f32(
                false, afrag[kk], false, bfrag[kk][1], (short)0, acc1, false, false);
        }

        // ---- scatter-add: D VGPR r -> row M = r + 8*half, col N = l16 (+16)
#pragma unroll
        for (int r = 0; r < 8; ++r) {
            const int m = r + half * 8;
            const int p = pbase + m;
            if (p < P) {
                const int oid = outik[p];
                float* orow = out + (size_t)oid * COUT;
                unsafeAtomicAdd(orow + l16,      acc0[r]);
                unsafeAtomicAdd(orow + l16 + 16, acc1[r]);
            }
        }
    }
}

// ---------------------------------------------------------------------------
extern "C" void kernel_launch(void* const* d_in, const int* in_sizes, int n_in,
                              void* d_out, int out_size, void* d_ws, size_t ws_size,
                              hipStream_t stream) {
    const float* feats   = (const float*)d_in[0];
    const float* weight  = (const float*)d_in[1];
    const float* bias    = (const float*)d_in[2];
    const int*   in_idx  = (const int*)d_in[3];
    const int*   out_idx = (const int*)d_in[4];
    float*       out     = (float*)d_out;

    const int K  = in_sizes[1] / (CIN * COUT);  // 27
    const int P  = in_sizes[3] / K;             // 50000
    const int total = out_size;                 // n_out * COUT

    bias_init_kernel<<<(total + 255) / 256, 256, 0, stream>>>(out, bias, total);

    dim3 grid((P + PAIRS_PER_BLOCK - 1) / PAIRS_PER_BLOCK, K);
    spconv_wmma_kernel<<<grid, 256, 0, stream>>>(feats, weight, in_idx, out_idx,
                                                 out, P);
}